// GNN_32693291057797
// MI455X (gfx1250) — compile-verified
//
#include <hip/hip_runtime.h>
#include <hip/hip_bf16.h>

typedef __attribute__((ext_vector_type(2))) float v2f;
typedef __attribute__((ext_vector_type(8))) float v8f;

#define HIDDEN 128

// ---------------------------------------------------------------- zero (f4)
__global__ void k_zero4(float4* __restrict__ p, long long n4) {
    long long i = (long long)blockIdx.x * blockDim.x + threadIdx.x;
    if (i < n4) p[i] = make_float4(0.f, 0.f, 0.f, 0.f);
}

// ------------------------------------------------- layer1 scatter (C_in = 2)
__global__ void k_scatter2(const float* __restrict__ x,
                           const int* __restrict__ src,
                           const int* __restrict__ dst,
                           const float* __restrict__ ew,
                           float* __restrict__ aggr, int E) {
    int e = blockIdx.x * blockDim.x + threadIdx.x;
    if (e >= E) return;
    int s = src[e], d = dst[e];
    float w = ew[e];
    atomicAdd(&aggr[(long long)d * 2 + 0], w * x[(long long)s * 2 + 0]);
    atomicAdd(&aggr[(long long)d * 2 + 1], w * x[(long long)s * 2 + 1]);
}

// --------------------------------------- layer1 dense: 2 -> 128 (+bias,relu)
// thread t handles node t/32, channels (t%32)*4 .. +3
__global__ void k_dense_l1(const float* __restrict__ x,
                           const float* __restrict__ aggr1,
                           const float* __restrict__ Wrel,   // [2,128]
                           const float* __restrict__ bias,   // [128]
                           const float* __restrict__ Wroot,  // [2,128]
                           float* __restrict__ h1, int N) {
    int t = blockIdx.x * blockDim.x + threadIdx.x;
    int node = t >> 5, chunk = t & 31;
    if (node >= N) return;
    float a0 = aggr1[(long long)node * 2 + 0];
    float a1 = aggr1[(long long)node * 2 + 1];
    float x0 = x[(long long)node * 2 + 0];
    float x1 = x[(long long)node * 2 + 1];
    int c = chunk * 4;
#pragma unroll
    for (int j = 0; j < 4; ++j) {
        int cc = c + j;
        float v = a0 * Wrel[cc] + a1 * Wrel[HIDDEN + cc] + bias[cc] +
                  x0 * Wroot[cc] + x1 * Wroot[HIDDEN + cc];
        h1[(long long)node * HIDDEN + cc] = fmaxf(v, 0.f);
    }
}

// ----------------------------------- generic 128-ch scatter: 32 lanes / edge
// lane handles 4 channels: coalesced float4 gather + 4 f32 atomics (L2-resident)
__global__ void k_scatter128(const float* __restrict__ h,
                             const int* __restrict__ src,
                             const int* __restrict__ dst,
                             const float* __restrict__ ew,
                             float* __restrict__ aggr, long long E) {
    long long t = (long long)blockIdx.x * blockDim.x + threadIdx.x;
    long long e = t >> 5;
    int chunk = (int)(t & 31);
    if (e >= E) return;
    int s = src[e], d = dst[e];
    float w = ew[e];
    const float4 v = *(const float4*)(h + (long long)s * HIDDEN + chunk * 4);
    float* base = aggr + (long long)d * HIDDEN + chunk * 4;
    atomicAdd(base + 0, w * v.x);
    atomicAdd(base + 1, w * v.y);
    atomicAdd(base + 2, w * v.z);
    atomicAdd(base + 3, w * v.w);
}

// ------------------- layer2 dense via fp32 WMMA: out = relu(A@Wr + b + H@Wq)
// 256 thr = 8 waves; wave w -> output-channel tile [w*16, w*16+16); block ->
// node tile of 16 rows. K loop: 32 x (16x16x4) WMMAs per weight matrix.
__global__ __launch_bounds__(256)
void k_dense128_wmma(const float* __restrict__ aggr,   // [N,128] "A" rel input
                     const float* __restrict__ h,      // [N,128] root input
                     const float* __restrict__ Wrel,   // [128,128]
                     const float* __restrict__ bias,   // [128]
                     const float* __restrict__ Wroot,  // [128,128]
                     float* __restrict__ out, int N) {
    const int wave = threadIdx.x >> 5;     // 0..7  -> N (out-channel) tile
    const int lane = threadIdx.x & 31;
    const int m     = lane & 15;           // M row within tile / N col for B,C
    const int khalf = lane >> 4;           // 0 or 1
    const int nodeBase = blockIdx.x * 16;
    const int col = wave * 16 + m;         // output channel

    if (nodeBase + 16 <= N) {
        const long long row = nodeBase + m;
        v8f acc = {};
        // A-frag (16x4 f32): lane m holds K = khalf*2 + {0,1} of row `row`
        // B-frag (4x16 f32): lane m holds W[khalf*2 + {0,1}][col]
        for (int k = 0; k < HIDDEN; k += 4) {
            const int ka = k + khalf * 2;
            v2f a_rel  = *(const v2f*)(aggr + row * HIDDEN + ka);
            v2f a_root = *(const v2f*)(h    + row * HIDDEN + ka);
            v2f b_rel, b_root;
            b_rel.x  = Wrel [(long long)ka * HIDDEN + col];
            b_rel.y  = Wrel [(long long)(ka + 1) * HIDDEN + col];
            b_root.x = Wroot[(long long)ka * HIDDEN + col];
            b_root.y = Wroot[(long long)(ka + 1) * HIDDEN + col];
            acc = __builtin_amdgcn_wmma_f32_16x16x4_f32(
                false, a_rel, false, b_rel, (short)0, acc, false, false);
            acc = __builtin_amdgcn_wmma_f32_16x16x4_f32(
                false, a_root, false, b_root, (short)0, acc, false, false);
        }
        // C/D layout: VGPR r, lanes 0-15 -> M=r, lanes 16-31 -> M=r+8; N=m
        const float bv = bias[col];
#pragma unroll
        for (int r = 0; r < 8; ++r) {
            const int mrow = nodeBase + r + khalf * 8;
            float v = acc[r] + bv;
            out[(long long)mrow * HIDDEN + col] = fmaxf(v, 0.f);
        }
    } else {
        // ragged tail (uniform per block): scalar fallback
        int c = threadIdx.x;
        if (c < HIDDEN) {
            for (int rrow = nodeBase; rrow < N; ++rrow) {
                float s = bias[c];
                for (int k = 0; k < HIDDEN; ++k) {
                    s += aggr[(long long)rrow * HIDDEN + k] * Wrel[(long long)k * HIDDEN + c];
                    s += h[(long long)rrow * HIDDEN + k] * Wroot[(long long)k * HIDDEN + c];
                }
                out[(long long)rrow * HIDDEN + c] = fmaxf(s, 0.f);
            }
        }
    }
}

// ------------------------- layer3 dense: 128 -> 3, wave-per-node + shfl reduce
__global__ void k_dense_l3(const float* __restrict__ aggr,
                           const float* __restrict__ h,
                           const float* __restrict__ Wrel,   // [128,3]
                           const float* __restrict__ bias,   // [3]
                           const float* __restrict__ Wroot,  // [128,3]
                           float* __restrict__ out, int N) {
    int lane = threadIdx.x & 31;
    int node = (blockIdx.x * blockDim.x + threadIdx.x) >> 5;
    if (node >= N) return;
    float p0 = 0.f, p1 = 0.f, p2 = 0.f;
    for (int k = lane; k < HIDDEN; k += 32) {
        float a  = aggr[(long long)node * HIDDEN + k];
        float hv = h[(long long)node * HIDDEN + k];
        p0 += a * Wrel[k * 3 + 0] + hv * Wroot[k * 3 + 0];
        p1 += a * Wrel[k * 3 + 1] + hv * Wroot[k * 3 + 1];
        p2 += a * Wrel[k * 3 + 2] + hv * Wroot[k * 3 + 2];
    }
#pragma unroll
    for (int off = 16; off > 0; off >>= 1) {
        p0 += __shfl_down(p0, off, 32);
        p1 += __shfl_down(p1, off, 32);
        p2 += __shfl_down(p2, off, 32);
    }
    if (lane == 0) {
        out[(long long)node * 3 + 0] = p0 + bias[0];
        out[(long long)node * 3 + 1] = p1 + bias[1];
        out[(long long)node * 3 + 2] = p2 + bias[2];
    }
}

// ---------------------------------------------------------------- launcher
extern "C" void kernel_launch(void* const* d_in, const int* in_sizes, int n_in,
                              void* d_out, int out_size, void* d_ws, size_t ws_size,
                              hipStream_t stream) {
    const float* x  = (const float*)d_in[0];
    const int*   ei = (const int*)d_in[1];     // [2,E] row-major
    const float* ew = (const float*)d_in[2];
    // d_in[3] = batch (unused)
    const float* W1rel  = (const float*)d_in[4];
    const float* b1     = (const float*)d_in[5];
    const float* W1root = (const float*)d_in[6];
    const float* W2rel  = (const float*)d_in[7];
    const float* b2     = (const float*)d_in[8];
    const float* W2root = (const float*)d_in[9];
    const float* W3rel  = (const float*)d_in[10];
    const float* b3     = (const float*)d_in[11];
    const float* W3root = (const float*)d_in[12];
    float* out = (float*)d_out;

    const int N = in_sizes[0] / 2;            // x is [N,2]
    const int E = in_sizes[2];                // edge_weight is [E]
    const int* src = ei;
    const int* dst = ei + E;

    // workspace partition (256B aligned)
    auto align256 = [](size_t v) { return (v + 255) & ~(size_t)255; };
    char* base = (char*)d_ws;
    size_t off = 0;
    float* aggr1 = (float*)(base + off); off = align256(off + (size_t)N * 2 * 4);
    float* h1    = (float*)(base + off); off = align256(off + (size_t)N * HIDDEN * 4);
    float* aggr2 = (float*)(base + off); off = align256(off + (size_t)N * HIDDEN * 4);
    float* h2    = (float*)(base + off); off = align256(off + (size_t)N * HIDDEN * 4);
    float* aggr3 = h1;  // h1 dead after layer-2 dense; reuse as aggr3
    (void)ws_size; (void)n_in; (void)out_size;

    const int B = 256;
    const long long nfeat  = (long long)N * HIDDEN;

    // 1) zero accumulators
    {
        long long n4 = ((long long)N * 2) / 4;
        k_zero4<<<(int)((n4 + B - 1) / B), B, 0, stream>>>((float4*)aggr1, n4);
        n4 = nfeat / 4;
        k_zero4<<<(int)((n4 + B - 1) / B), B, 0, stream>>>((float4*)aggr2, n4);
    }
    // 2) layer 1
    k_scatter2<<<(E + B - 1) / B, B, 0, stream>>>(x, src, dst, ew, aggr1, E);
    k_dense_l1<<<(int)(((long long)N * 32 + B - 1) / B), B, 0, stream>>>(
        x, aggr1, W1rel, b1, W1root, h1, N);
    // 3) layer 2
    {
        long long t = (long long)E * 32;
        k_scatter128<<<(int)((t + B - 1) / B), B, 0, stream>>>(h1, src, dst, ew, aggr2, E);
    }
    k_dense128_wmma<<<(N + 15) / 16, 256, 0, stream>>>(aggr2, h1, W2rel, b2, W2root, h2, N);
    // 4) layer 3
    {
        long long n4 = nfeat / 4;
        k_zero4<<<(int)((n4 + B - 1) / B), B, 0, stream>>>((float4*)aggr3, n4);
        long long t = (long long)E * 32;
        k_scatter128<<<(int)((t + B - 1) / B), B, 0, stream>>>(h2, src, dst, ew, aggr3, E);
    }
    k_dense_l3<<<(int)(((long long)N * 32 + B - 1) / B), B, 0, stream>>>(
        aggr3, h2, W3rel, b3, W3root, out, N);
}